// Prop_36472862278040
// MI455X (gfx1250) — compile-verified
//
#include <hip/hip_runtime.h>

typedef float v2f __attribute__((ext_vector_type(2)));
typedef float v8f __attribute__((ext_vector_type(8)));

#define DF 64
#define SCAN_B 256

// ---------------- CSR build helpers ----------------

__global__ void zero_i32_kernel(int* __restrict__ p, long long n) {
  long long i = (long long)blockIdx.x * blockDim.x + threadIdx.x;
  if (i < n) p[i] = 0;
}

__global__ void hist_kernel(const int* __restrict__ dst, int* __restrict__ counts, int e) {
  int i = blockIdx.x * blockDim.x + threadIdx.x;
  if (i < e) atomicAdd(&counts[dst[i]], 1);
}

// Per-block exclusive scan of vals -> out (local), block totals -> blockSums.
__global__ void scan1_kernel(const int* __restrict__ vals, int* __restrict__ out,
                             int* __restrict__ blockSums, int n) {
  __shared__ int tmp[SCAN_B];
  int g = blockIdx.x * SCAN_B + threadIdx.x;
  int v = (g < n) ? vals[g] : 0;
  tmp[threadIdx.x] = v;
  __syncthreads();
  for (int d = 1; d < SCAN_B; d <<= 1) {
    int t = (threadIdx.x >= (unsigned)d) ? tmp[threadIdx.x - d] : 0;
    __syncthreads();
    tmp[threadIdx.x] += t;
    __syncthreads();
  }
  if (g < n) out[g] = tmp[threadIdx.x] - v;  // exclusive
  if (threadIdx.x == SCAN_B - 1) blockSums[blockIdx.x] = tmp[threadIdx.x];
}

// Serial scan of block sums (tiny).
__global__ void scan2_kernel(const int* __restrict__ blockSums, int* __restrict__ blockOffs, int nb) {
  if (blockIdx.x == 0 && threadIdx.x == 0) {
    int run = 0;
    for (int i = 0; i < nb; ++i) { blockOffs[i] = run; run += blockSums[i]; }
    blockOffs[nb] = run;
  }
}

__global__ void scan3_kernel(int* __restrict__ out, const int* __restrict__ blockOffs, int n) {
  int g = blockIdx.x * SCAN_B + threadIdx.x;
  if (g < n) out[g] += blockOffs[blockIdx.x];
  if (g == 0) out[n] = blockOffs[gridDim.x];  // grand total
}

// Per-tile edge count rounded up to a multiple of 8 (branch-free inner loop).
__global__ void tilecnt_kernel(const int* __restrict__ offs, int* __restrict__ tcnt, int tiles) {
  int t = blockIdx.x * blockDim.x + threadIdx.x;
  if (t < tiles) tcnt[t] = ((offs[t * 16 + 16] - offs[t * 16]) + 7) & ~7;
}

// Scatter edges into padded per-tile CSR; pack (src | row<<20, w) into 8 bytes.
// Padding slots stay zero => src=0,row=0,w=0 => contribute nothing.
__global__ void scatter_kernel(const int* __restrict__ src, const int* __restrict__ dst,
                               const float* __restrict__ w, const int* __restrict__ offs,
                               const int* __restrict__ tileOffs, int* __restrict__ cursor,
                               uint2* __restrict__ e_pack, int e) {
  int i = blockIdx.x * blockDim.x + threadIdx.x;
  if (i < e) {
    int d = dst[i];
    int t = d >> 4;
    int rank = atomicAdd(&cursor[d], 1);
    int pos = tileOffs[t] + (offs[d] - offs[t * 16]) + rank;
    uint2 v;
    v.x = (unsigned)src[i] | ((unsigned)(d & 15) << 20);
    v.y = __float_as_uint(w[i]);
    e_pack[pos] = v;
  }
}

// ---------------- WMMA propagation ----------------
// One wave per tile of 16 dst nodes. 4 edges per WMMA step:
//   A(16x4) selection matrix: A[i][j] = w_j iff edge j's dst-row == i
//   B(4x16) gathered h[src_j][feature slice]
//   C += A x B  == weighted segment-sum for 16 nodes x 16 features.
// Metadata for this lane-half's two edges comes in as ONE b128 load
// (edges eA, eA+1 are contiguous uint2's, 16B aligned since eA is even).
// Weight select is a bitwise and-mask so no load can be sunk under a branch.
__device__ __forceinline__ void chunk4(const uint4* __restrict__ ep2, int p, int hlf, int m,
                                       const float* __restrict__ hin,
                                       v8f& c0, v8f& c1, v8f& c2, v8f& c3) {
  const uint4 md = ep2[(p >> 1) + hlf];   // {packA, wbitsA, packB, wbitsB}
  const int sA = (int)(md.x & 0xFFFFFu);
  const int sB = (int)(md.z & 0xFFFFFu);
  const unsigned mskA = ((md.x >> 20) == (unsigned)m) ? 0xFFFFFFFFu : 0u;
  const unsigned mskB = ((md.z >> 20) == (unsigned)m) ? 0xFFFFFFFFu : 0u;

  v2f a;
  a[0] = __uint_as_float(md.y & mskA);
  a[1] = __uint_as_float(md.w & mskB);

  const float* __restrict__ rowA = hin + (size_t)sA * DF;
  const float* __restrict__ rowB = hin + (size_t)sB * DF;

  v2f b0, b1, b2, b3;
  b0[0] = rowA[m];      b0[1] = rowB[m];
  b1[0] = rowA[16 + m]; b1[1] = rowB[16 + m];
  b2[0] = rowA[32 + m]; b2[1] = rowB[32 + m];
  b3[0] = rowA[48 + m]; b3[1] = rowB[48 + m];

  c0 = __builtin_amdgcn_wmma_f32_16x16x4_f32(false, a, false, b0, (short)0, c0, false, false);
  c1 = __builtin_amdgcn_wmma_f32_16x16x4_f32(false, a, false, b1, (short)0, c1, false, false);
  c2 = __builtin_amdgcn_wmma_f32_16x16x4_f32(false, a, false, b2, (short)0, c2, false, false);
  c3 = __builtin_amdgcn_wmma_f32_16x16x4_f32(false, a, false, b3, (short)0, c3, false, false);
}

__global__ __launch_bounds__(32) void prop_wmma_kernel(
    const float* __restrict__ hin, float* __restrict__ hout,
    const int* __restrict__ tileOffs, const uint4* __restrict__ e_pack2, int n) {
  const int tile = blockIdx.x;
  const int lane = threadIdx.x;
  const int hlf  = lane >> 4;
  const int m    = lane & 15;

  const int begin = tileOffs[tile];
  const int end   = tileOffs[tile + 1];   // end - begin is a multiple of 8

  v8f c0 = {}, c1 = {}, c2 = {}, c3 = {};

  for (int p = begin; p < end; p += 8) {
    chunk4(e_pack2, p,     hlf, m, hin, c0, c1, c2, c3);
    chunk4(e_pack2, p + 4, hlf, m, hin, c0, c1, c2, c3);
  }

  // D layout: VGPR r -> row (r + 8*hlf), col (lane&15)
  const int nodeBase = tile * 16;
  if (nodeBase + 16 <= n) {               // uniform: full tile (always true when n%16==0)
#pragma unroll
    for (int r = 0; r < 8; ++r) {
      float* __restrict__ o = hout + (size_t)(nodeBase + r + hlf * 8) * DF + m;
      o[0]  = c0[r];
      o[16] = c1[r];
      o[32] = c2[r];
      o[48] = c3[r];
    }
  } else {
#pragma unroll
    for (int r = 0; r < 8; ++r) {
      const int row = nodeBase + r + hlf * 8;
      if (row < n) {
        float* __restrict__ o = hout + (size_t)row * DF + m;
        o[0]  = c0[r];
        o[16] = c1[r];
        o[32] = c2[r];
        o[48] = c3[r];
      }
    }
  }
}

// ---------------- launch ----------------

extern "C" void kernel_launch(void* const* d_in, const int* in_sizes, int n_in,
                              void* d_out, int out_size, void* d_ws, size_t ws_size,
                              hipStream_t stream) {
  const float* x   = (const float*)d_in[0];
  const float* ew  = (const float*)d_in[1];
  const int*   src = (const int*)d_in[2];
  const int*   dst = (const int*)d_in[3];

  const int N     = in_sizes[0] / DF;   // 100000 (multiple of 16)
  const int E     = in_sizes[1];        // 1600000
  const int nb    = (N + SCAN_B - 1) / SCAN_B;
  const int tiles = (N + 15) / 16;
  const int nbT   = (tiles + SCAN_B - 1) / SCAN_B;
  const long long packCap = (long long)E + 8LL * tiles;   // padded edge capacity

  char* p = (char*)d_ws;
  auto take = [&](size_t bytes) -> char* {
    char* q = p;
    p += (bytes + 255) & ~(size_t)255;
    return q;
  };
  int*   counts    = (int*)take((size_t)N * 4);            // reused as scatter cursor
  int*   offs      = (int*)take((size_t)(N + 1) * 4);
  int*   blockSums = (int*)take((size_t)(nb + 1) * 4);     // reused for tile scan (nbT << nb)
  int*   blockOffs = (int*)take((size_t)(nb + 1) * 4);
  int*   tcnt      = (int*)take((size_t)tiles * 4);
  int*   tileOffs  = (int*)take((size_t)(tiles + 1) * 4);
  uint2* e_pack    = (uint2*)take((size_t)packCap * 8);
  float* bufA      = (float*)take((size_t)N * DF * 4);
  (void)ws_size; (void)n_in; (void)out_size;

  // ---- one-time padded-CSR build (int atomics only) ----
  zero_i32_kernel<<<(N + 255) / 256, 256, 0, stream>>>(counts, N);
  hist_kernel<<<(E + 255) / 256, 256, 0, stream>>>(dst, counts, E);
  scan1_kernel<<<nb, SCAN_B, 0, stream>>>(counts, offs, blockSums, N);
  scan2_kernel<<<1, 1, 0, stream>>>(blockSums, blockOffs, nb);
  scan3_kernel<<<nb, SCAN_B, 0, stream>>>(offs, blockOffs, N);

  tilecnt_kernel<<<(tiles + 255) / 256, 256, 0, stream>>>(offs, tcnt, tiles);
  scan1_kernel<<<nbT, SCAN_B, 0, stream>>>(tcnt, tileOffs, blockSums, tiles);
  scan2_kernel<<<1, 1, 0, stream>>>(blockSums, blockOffs, nbT);
  scan3_kernel<<<nbT, SCAN_B, 0, stream>>>(tileOffs, blockOffs, tiles);

  zero_i32_kernel<<<(N + 255) / 256, 256, 0, stream>>>(counts, N);  // cursor
  const long long packInts = packCap * 2;
  zero_i32_kernel<<<(int)((packInts + 255) / 256), 256, 0, stream>>>((int*)e_pack, packInts);
  scatter_kernel<<<(E + 255) / 256, 256, 0, stream>>>(src, dst, ew, offs, tileOffs,
                                                      counts, e_pack, E);

  // ---- K=8 propagation: x -> bufA -> d_out -> bufA -> ... -> d_out ----
  float* out = (float*)d_out;
  const float* hin = x;
  for (int k = 0; k < 8; ++k) {
    float* o = ((k & 1) == 0) ? bufA : out;
    prop_wmma_kernel<<<tiles, 32, 0, stream>>>(hin, o, tileOffs, (const uint4*)e_pack, N);
    hin = o;
  }
}